// residual_attn_block_39496519254368
// MI455X (gfx1250) — compile-verified
//
#include <hip/hip_runtime.h>
#include <hip/hip_bf16.h>

// ---------------------------------------------------------------------------
// Problem constants (from reference): B=32, C=64, P=1024, K=20, CO=64, CMV=1024
// ---------------------------------------------------------------------------
#define BB   32
#define CC   64
#define PP   1024
#define KNN  20
#define COO  64
#define CMV  1024
#define WPC_COLS 128     // 2*C
#define WMV_COLS 1088    // C + CMV
#define BN_EPS 1e-5f

typedef __attribute__((ext_vector_type(2))) float v2f;
typedef __attribute__((ext_vector_type(8))) float v8f;

// D = A(16x4 f32) * B(4x16 f32) + C(16x16 f32)   -> v_wmma_f32_16x16x4_f32
__device__ __forceinline__ v8f wmma_f32_4(v2f a, v2f b, v8f c) {
  return __builtin_amdgcn_wmma_f32_16x16x4_f32(
      /*neg_a=*/false, a, /*neg_b=*/false, b,
      /*c_mod=*/(short)0, c, /*reuse_a=*/false, /*reuse_b=*/false);
}

// ---------------------------------------------------------------------------
// Kernel 1: per (b, 16-point tile) compute the full 16x1024 adj row-block via
// f32 WMMA syrk into LDS in ONE pass (afrag registers live only here), then
// run register-resident top-20 per row (bd/bi live only here) -> disjoint
// register lifetimes, no scratch spills in the WMMA loop.
// Block = 128 threads (4 waves).
// LDS: sq 4KB + adj 16*1044*4 = 66.8KB  (CDNA5 allows up to 320KB per WG).
// ---------------------------------------------------------------------------
#define ADJ_S 1044   // 1024 + 20 pad: 1044 % 64 == 20 -> conflict-free row scans

__global__ void knn_kernel(const float* __restrict__ x, int* __restrict__ nn_idx) {
  const int b    = blockIdx.y;
  const int pt   = blockIdx.x * 16;
  const int tid  = threadIdx.x;
  const int lane = tid & 31;
  const int wave = tid >> 5;

  __shared__ float sq[PP];
  __shared__ float adj[16 * ADJ_S];

  const float* xb = x + (size_t)b * CC * PP;

  // |x_q|^2 for all q
  for (int q = tid; q < PP; q += 128) {
    float s = 0.f;
    #pragma unroll 8
    for (int c = 0; c < CC; ++c) { float v = xb[c * PP + q]; s += v * v; }
    sq[q] = s;
  }
  __syncthreads();

  const int mrow  = lane & 15;         // M (or N) index held by this lane
  const int khi   = (lane >> 4) * 2;   // K sub-offset for this half-wave
  const int rbase = (lane >> 4) * 8;   // C/D row base for this half-wave

  // Preload the 16 A-fragments for the p-tile (loop invariant).
  v2f afrag[16];
  #pragma unroll
  for (int i = 0; i < 16; ++i) {
    const int c0 = i * 4 + khi;
    afrag[i].x = xb[(c0 + 0) * PP + pt + mrow];
    afrag[i].y = xb[(c0 + 1) * PP + pt + mrow];
  }
  float sqp[8];
  #pragma unroll
  for (int r = 0; r < 8; ++r) sqp[r] = sq[pt + rbase + r];

  // 64 q-tiles of 16, round-robin over 4 waves
  for (int jt = wave; jt < 64; jt += 4) {
    const int qb = jt * 16;
    v8f acc = {0.f, 0.f, 0.f, 0.f, 0.f, 0.f, 0.f, 0.f};
    #pragma unroll
    for (int i = 0; i < 16; ++i) {
      const int c0 = i * 4 + khi;
      v2f bf;
      bf.x = xb[(c0 + 0) * PP + qb + mrow];
      bf.y = xb[(c0 + 1) * PP + qb + mrow];
      acc = wmma_f32_4(afrag[i], bf, acc);   // inner[p_tile, q_tile]
    }
    const float sqq = sq[qb + mrow];
    #pragma unroll
    for (int r = 0; r < 8; ++r) {
      adj[(rbase + r) * ADJ_S + qb + mrow] = sqp[r] + sqq - 2.f * acc[r];
    }
  }
  __syncthreads();

  // 16 threads, one row each: top-20 smallest via unrolled register insertion
  if (tid < 16) {
    float bd[KNN]; int bi[KNN];
    #pragma unroll
    for (int j = 0; j < KNN; ++j) { bd[j] = 3.4e38f; bi[j] = 0; }
    const float* arow = &adj[tid * ADJ_S];
    for (int qq = 0; qq < PP; ++qq) {
      const float d = arow[qq];
      if (d < bd[KNN - 1]) {
        bd[KNN - 1] = d; bi[KNN - 1] = qq;
        #pragma unroll
        for (int j = KNN - 1; j > 0; --j) {
          if (bd[j] < bd[j - 1]) {
            float td = bd[j]; bd[j] = bd[j - 1]; bd[j - 1] = td;
            int   ti = bi[j]; bi[j] = bi[j - 1]; bi[j - 1] = ti;
          }
        }
      }
    }
    int* out = nn_idx + ((size_t)b * PP + pt + tid) * KNN;
    #pragma unroll
    for (int j = 0; j < KNN; ++j) out[j] = bi[j];
  }
}

// ---------------------------------------------------------------------------
// Kernel 2: edge conv. Per (b, 4-point group): stage E (128 x 80) in LDS,
// GEMM W_pc(64x128) x E via f32 WMMA, fused BN1+ReLU (per-channel affine
// precomputed in LDS), max over K via ds_max_u32 (float bits are
// order-preserving for non-negative values).
// Block = 256 threads (8 waves). LDS: E 42KB + pmax 1KB + affine 0.5KB.
// ---------------------------------------------------------------------------
#define PTS  4
#define COLS 80          // PTS * KNN = 5 N-tiles of 16
#define E_S  84          // padded row stride

__global__ void edgeconv_kernel(const float* __restrict__ x,
                                const int*   __restrict__ nn_idx,
                                const float* __restrict__ w_pc,
                                const float* __restrict__ b_pc,
                                const float* __restrict__ g1,
                                const float* __restrict__ be1,
                                const float* __restrict__ m1,
                                const float* __restrict__ v1,
                                float* __restrict__ pc_out) {
  const int b     = blockIdx.y;
  const int pbase = blockIdx.x * PTS;
  const int tid   = threadIdx.x;
  const int lane  = tid & 31;
  const int wave  = tid >> 5;

  __shared__ float    E[WPC_COLS * E_S];
  __shared__ unsigned pmax[COO * PTS];
  __shared__ float    alpha[COO], beta[COO];

  const float* xb = x + (size_t)b * CC * PP;

  for (int i = tid; i < COO * PTS; i += 256) pmax[i] = 0u;
  if (tid < COO) {
    // y = (acc + b_pc - m1)*inv + be1  ==  acc*alpha + beta
    const float inv = g1[tid] * rsqrtf(v1[tid] + BN_EPS);
    alpha[tid] = inv;
    beta[tid]  = (b_pc[tid] - m1[tid]) * inv + be1[tid];
  }

  // Stage edge matrix: E[c][p_local*K + k]
  for (int idx = tid; idx < WPC_COLS * COLS; idx += 256) {
    const int c   = idx / COLS;
    const int col = idx - c * COLS;
    const int pl  = col / KNN;
    const int k   = col - pl * KNN;
    const int p   = pbase + pl;
    float val;
    if (c < CC) {
      val = xb[c * PP + p];                       // broadcast center
    } else {
      const int q = nn_idx[((size_t)b * PP + p) * KNN + k];
      val = xb[(c - CC) * PP + q] - xb[(c - CC) * PP + p];   // neigh - center
    }
    E[c * E_S + col] = val;
  }
  __syncthreads();

  const int mrow = lane & 15;
  const int khi  = (lane >> 4) * 2;

  // 4 o-tiles x 5 n-tiles = 20 output tiles, round-robin over 8 waves
  for (int t = wave; t < 20; t += 8) {
    const int ot    = t / 5;
    const int nt    = t - ot * 5;
    const int obase = ot * 16;
    const int cb    = nt * 16;
    v8f acc = {0.f, 0.f, 0.f, 0.f, 0.f, 0.f, 0.f, 0.f};
    #pragma unroll
    for (int i = 0; i < 32; ++i) {               // contract over 128 channels
      const int c0 = i * 4 + khi;
      v2f a, bf;
      a.x  = w_pc[(obase + mrow) * WPC_COLS + c0 + 0];
      a.y  = w_pc[(obase + mrow) * WPC_COLS + c0 + 1];
      bf.x = E[(c0 + 0) * E_S + cb + mrow];
      bf.y = E[(c0 + 1) * E_S + cb + mrow];
      acc = wmma_f32_4(a, bf, acc);
    }
    const int rbase = (lane >> 4) * 8;
    const int col   = cb + mrow;
    const int pl    = col / KNN;
    #pragma unroll
    for (int r = 0; r < 8; ++r) {
      const int o = obase + rbase + r;
      float y = fmaxf(acc[r] * alpha[o] + beta[o], 0.f);
      atomicMax(&pmax[o * PTS + pl], __float_as_uint(y));  // ds_max_u32
    }
  }
  __syncthreads();

  for (int i = tid; i < COO * PTS; i += 256) {
    const int o  = i / PTS;
    const int pl = i - o * PTS;
    pc_out[(size_t)b * COO * PP + (size_t)o * PP + pbase + pl] =
        __uint_as_float(pmax[i]);
  }
}

// ---------------------------------------------------------------------------
// Kernel 3a: base[b,o] = b_mv[o] + W_mv[o, :1024] . mv_ft[b]   (mv channels
// are constant over p, so this term is hoisted out of the per-point GEMM)
// ---------------------------------------------------------------------------
__global__ void mv_base_kernel(const float* __restrict__ w_mv,
                               const float* __restrict__ b_mv,
                               const float* __restrict__ mv_ft,
                               float* __restrict__ base) {
  const int b = blockIdx.x;
  const int o = threadIdx.x;
  const float* w = w_mv  + (size_t)o * WMV_COLS;
  const float* f = mv_ft + (size_t)b * CMV;
  float s = b_mv[o];
  #pragma unroll 8
  for (int c = 0; c < CMV; ++c) s += w[c] * f[c];
  base[b * COO + o] = s;
}

// ---------------------------------------------------------------------------
// Kernel 3b: mask GEMM  W_mv[:,1024:1088] (64x64) x x[b](64xP)  via f32 WMMA,
// fused epilogue: + base, BN2, ReLU, sigmoid(log y)=y/(1+y), out=(m+1)*pc_out.
// Block = 128 threads (4 waves = 4 o-tiles), grid (P/16, B).
// ---------------------------------------------------------------------------
__global__ void mask_out_kernel(const float* __restrict__ x,
                                const float* __restrict__ w_mv,
                                const float* __restrict__ base,
                                const float* __restrict__ g2,
                                const float* __restrict__ be2,
                                const float* __restrict__ m2,
                                const float* __restrict__ v2,
                                const float* __restrict__ pc_out,
                                float* __restrict__ out) {
  const int b     = blockIdx.y;
  const int pb    = blockIdx.x * 16;
  const int lane  = threadIdx.x & 31;
  const int obase = (threadIdx.x >> 5) * 16;
  const int mrow  = lane & 15;
  const int khi   = (lane >> 4) * 2;
  const float* xb = x + (size_t)b * CC * PP;

  v8f acc = {0.f, 0.f, 0.f, 0.f, 0.f, 0.f, 0.f, 0.f};
  #pragma unroll
  for (int i = 0; i < 16; ++i) {                 // contract over 64 x-channels
    const int c0 = i * 4 + khi;
    v2f a, bf;
    a.x  = w_mv[(obase + mrow) * WMV_COLS + CMV + c0 + 0];
    a.y  = w_mv[(obase + mrow) * WMV_COLS + CMV + c0 + 1];
    bf.x = xb[(c0 + 0) * PP + pb + mrow];
    bf.y = xb[(c0 + 1) * PP + pb + mrow];
    acc = wmma_f32_4(a, bf, acc);
  }
  const int rbase = (lane >> 4) * 8;
  const int p = pb + mrow;
  #pragma unroll
  for (int r = 0; r < 8; ++r) {
    const int o = obase + rbase + r;
    float y = acc[r] + base[b * COO + o];
    const float inv = g2[o] * rsqrtf(v2[o] + BN_EPS);
    y = (y - m2[o]) * inv + be2[o];
    y = fmaxf(y, 0.f);
    const float m = y / (1.f + y);               // sigmoid(log(relu(y)))
    const size_t oi = (size_t)b * COO * PP + (size_t)o * PP + p;
    const float po = pc_out[oi];
    out[oi] = m * po + po;
  }
}

// ---------------------------------------------------------------------------
// Launch
// ---------------------------------------------------------------------------
extern "C" void kernel_launch(void* const* d_in, const int* in_sizes, int n_in,
                              void* d_out, int out_size, void* d_ws, size_t ws_size,
                              hipStream_t stream) {
  (void)in_sizes; (void)n_in; (void)out_size; (void)ws_size;

  const float* pc    = (const float*)d_in[0];
  const float* mv_ft = (const float*)d_in[1];
  const float* w_pc  = (const float*)d_in[2];
  const float* b_pc  = (const float*)d_in[3];
  const float* g1    = (const float*)d_in[4];
  const float* be1   = (const float*)d_in[5];
  const float* m1    = (const float*)d_in[6];
  const float* v1    = (const float*)d_in[7];
  const float* w_mv  = (const float*)d_in[8];
  const float* b_mv  = (const float*)d_in[9];
  const float* g2    = (const float*)d_in[10];
  const float* be2   = (const float*)d_in[11];
  const float* m2    = (const float*)d_in[12];
  const float* v2    = (const float*)d_in[13];
  float* out = (float*)d_out;

  // Workspace layout (≈11 MB): nn_idx | pc_out | base
  char* ws = (char*)d_ws;
  int*   nn_idx = (int*)ws;                                   // 32*1024*20*4 = 2,621,440 B
  float* pc_out = (float*)(ws + 2621440);                     // 32*64*1024*4 = 8,388,608 B
  float* base   = (float*)(ws + 2621440 + 8388608);           // 32*64*4      = 8,192 B

  knn_kernel     <<<dim3(PP / 16,  BB), 128, 0, stream>>>(pc, nn_idx);
  edgeconv_kernel<<<dim3(PP / PTS, BB), 256, 0, stream>>>(pc, nn_idx, w_pc, b_pc,
                                                          g1, be1, m1, v1, pc_out);
  mv_base_kernel <<<BB, COO, 0, stream>>>(w_mv, b_mv, mv_ft, base);
  mask_out_kernel<<<dim3(PP / 16,  BB), 128, 0, stream>>>(pc, w_mv, base,
                                                          g2, be2, m2, v2, pc_out, out);
}